// AttentiveStatisticsPooling_22677427323330
// MI455X (gfx1250) — compile-verified
//
#include <hip/hip_runtime.h>
#include <hip/hip_bf16.h>
#include <math.h>

// ---------------------------------------------------------------------------
// AttentiveStatisticsPooling for MI455X (gfx1250, wave32)
//
// x:[32,1536,2000] f32 (393MB). Two unavoidable full passes over x ->
// ~786MB / 23.3 TB/s ~= 34us bandwidth floor; compute is ~1 GFLOP (noise).
// Both passes are f32 matvecs, implemented on V_WMMA_F32_16X16X4_F32 (exact
// f32 MACC) with the vector operand broadcast across all 16 B-columns; every
// column of D holds the identical matvec result, extracted from lanes 0/16.
// ---------------------------------------------------------------------------

typedef __attribute__((ext_vector_type(2))) float v2f;
typedef __attribute__((ext_vector_type(8))) float v8f;

#define NB 32
#define NC 1536
#define NT 2000
#define BN_EPS 1e-5f

// ws (float) layout:
//   [0] sum_a   [1] sum_a2   [2] mu   [3] scale = gamma*rsqrt(var+eps)
//   [8              .. 8 +   NB*NT) : a    (attention logits)
//   [8 + NB*NT      .. 8 + 2*NB*NT) : attn (softmax weights)
#define WS_A 8
#define WS_ATTN (8 + NB * NT)

__global__ void k_init(float* __restrict__ ws) {
    ws[0] = 0.0f;
    ws[1] = 0.0f;
}

// ---------------------------------------------------------------------------
// Kernel 1: a[b,t] = sum_c x[b,c,t]*w[c] + cb, plus global sum / sum-of-sq.
// One wave per (b, 16-wide t tile). A[m=t,k=c] (16x4), B[k,n]=w[c0+k] bcast.
// 4000 waves, 384 WMMAs each. A-loads: 16 consecutive t floats per c-row.
// ---------------------------------------------------------------------------
__global__ void __launch_bounds__(256)
k_attn_logits(const float* __restrict__ x, const float* __restrict__ w,
              const float* __restrict__ cb, float* __restrict__ ws) {
    const int lane = threadIdx.x & 31;
    const int wave = (int)((blockIdx.x * blockDim.x + threadIdx.x) >> 5);
    const int tilesPerB = NT / 16;              // 125
    const int b  = wave / tilesPerB;
    const int t0 = (wave % tilesPerB) * 16;
    const int m  = lane & 15;                   // A row -> t offset
    const int kb = (lane >> 4) << 1;            // K base: 0 or 2

    const float* __restrict__ xb = x + (size_t)b * NC * NT;

    v8f acc = {};
    for (int c0 = 0; c0 < NC; c0 += 4) {
        v2f av, bv;
        av.x = xb[(size_t)(c0 + kb    ) * NT + t0 + m];
        av.y = xb[(size_t)(c0 + kb + 1) * NT + t0 + m];
        bv.x = w[c0 + kb];
        bv.y = w[c0 + kb + 1];
        acc = __builtin_amdgcn_wmma_f32_16x16x4_f32(
            false, av, false, bv, (short)0, acc, false, false);
    }

    // All 16 columns of D are identical. Lane 0 holds rows 0..7 (t0+0..7),
    // lane 16 holds rows 8..15 (t0+8..15).
    if ((lane & 15) == 0) {
        const float cbv   = cb[0];
        const int   rbase = (lane >> 4) * 8;
        float* __restrict__ a_out = ws + WS_A + (size_t)b * NT;
        float s1 = 0.0f, s2 = 0.0f;
#pragma unroll
        for (int r = 0; r < 8; ++r) {
            float v = acc[r] + cbv;
            a_out[t0 + rbase + r] = v;
            s1 += v;
            s2 += v * v;
        }
        atomicAdd(&ws[0], s1);
        atomicAdd(&ws[1], s2);
    }
}

// ---------------------------------------------------------------------------
// Kernel 2: finalize BatchNorm batch statistics (biased variance).
// ---------------------------------------------------------------------------
__global__ void k_bn_finalize(float* __restrict__ ws, const float* __restrict__ gamma) {
    const float n   = (float)(NB * NT);
    const float mu  = ws[0] / n;
    const float var = ws[1] / n - mu * mu;
    ws[2] = mu;
    ws[3] = gamma[0] * rsqrtf(var + BN_EPS);
}

// ---------------------------------------------------------------------------
// Kernel 3: attn[b,:] = softmax_t( tanh( (a-mu)*scale + beta ) ). Block per b.
// ---------------------------------------------------------------------------
__global__ void __launch_bounds__(256)
k_softmax(float* __restrict__ ws, const float* __restrict__ beta) {
    __shared__ float red[256];
    const int   b   = blockIdx.x;
    const int   tid = threadIdx.x;
    const float mu  = ws[2];
    const float sc  = ws[3];
    const float bb  = beta[0];
    const float* __restrict__ a  = ws + WS_A    + (size_t)b * NT;
    float*       __restrict__ at = ws + WS_ATTN + (size_t)b * NT;

    float mx = -1e30f;
    for (int t = tid; t < NT; t += 256) {
        float z = tanhf((a[t] - mu) * sc + bb);
        at[t] = z;
        mx = fmaxf(mx, z);
    }
    red[tid] = mx;
    __syncthreads();
    for (int o = 128; o > 0; o >>= 1) {
        if (tid < o) red[tid] = fmaxf(red[tid], red[tid + o]);
        __syncthreads();
    }
    mx = red[0];
    __syncthreads();

    float sum = 0.0f;
    for (int t = tid; t < NT; t += 256) {
        float e = expf(at[t] - mx);
        at[t] = e;
        sum += e;
    }
    red[tid] = sum;
    __syncthreads();
    for (int o = 128; o > 0; o >>= 1) {
        if (tid < o) red[tid] += red[tid + o];
        __syncthreads();
    }
    const float inv = 1.0f / red[0];
    for (int t = tid; t < NT; t += 256) at[t] *= inv;
}

// ---------------------------------------------------------------------------
// Kernel 4: weighted statistics pooling.
//   mean[b,c] = sum_t x*attn ; E2[b,c] = sum_t x^2*attn ;
//   std = sqrt(max(E2 - mean^2, 1e-10))
// One wave per (b, 16-channel tile). A[m=c,k=t]: each lane loads a float2 of
// consecutive t from its channel row (each wave streams 16 rows sequentially).
// B[k,n] = attn[t0+k] broadcast. Two WMMA chains (x and x^2). 3072 waves.
// ---------------------------------------------------------------------------
__global__ void __launch_bounds__(256)
k_pool(const float* __restrict__ x, const float* __restrict__ ws,
       float* __restrict__ out) {
    const int lane = threadIdx.x & 31;
    const int wave = (int)((blockIdx.x * blockDim.x + threadIdx.x) >> 5);
    const int tilesPerB = NC / 16;              // 96
    const int b  = wave / tilesPerB;
    const int c0 = (wave % tilesPerB) * 16;
    const int m  = lane & 15;                   // A row -> channel offset
    const int kb = (lane >> 4) << 1;            // K base: 0 or 2

    const float* __restrict__ xr = x + ((size_t)b * NC + (c0 + m)) * NT;
    const float* __restrict__ at = ws + WS_ATTN + (size_t)b * NT;

    v8f accm = {};
    v8f accs = {};
    for (int t0 = 0; t0 < NT; t0 += 4) {
        // stream ~1KB ahead on this wave's channel row
        __builtin_prefetch(xr + t0 + kb + 256, 0, 1);
        v2f av = *(const v2f*)(xr + t0 + kb);   // 8B-aligned (NT, kb even)
        v2f bv = *(const v2f*)(at + t0 + kb);
        v2f a2;
        a2.x = av.x * av.x;
        a2.y = av.y * av.y;
        accm = __builtin_amdgcn_wmma_f32_16x16x4_f32(
            false, av, false, bv, (short)0, accm, false, false);
        accs = __builtin_amdgcn_wmma_f32_16x16x4_f32(
            false, a2, false, bv, (short)0, accs, false, false);
    }

    // Lane 0: channels c0+0..7 ; lane 16: channels c0+8..15.
    if ((lane & 15) == 0) {
        const int rbase = (lane >> 4) * 8;
        float* __restrict__ ob = out + (size_t)b * (2 * NC);
#pragma unroll
        for (int r = 0; r < 8; ++r) {
            float mean = accm[r];
            float var  = accs[r] - mean * mean;
            var = fmaxf(var, 1e-10f);
            ob[c0 + rbase + r]      = mean;
            ob[NC + c0 + rbase + r] = sqrtf(var);
        }
    }
}

extern "C" void kernel_launch(void* const* d_in, const int* in_sizes, int n_in,
                              void* d_out, int out_size, void* d_ws, size_t ws_size,
                              hipStream_t stream) {
    (void)in_sizes; (void)n_in; (void)out_size; (void)ws_size;
    const float* x      = (const float*)d_in[0];
    const float* conv_w = (const float*)d_in[1];
    const float* conv_b = (const float*)d_in[2];
    const float* gamma  = (const float*)d_in[3];
    const float* beta   = (const float*)d_in[4];
    float* out = (float*)d_out;
    float* ws  = (float*)d_ws;   // needs (8 + 2*NB*NT)*4 = ~512KB

    k_init<<<dim3(1), dim3(1), 0, stream>>>(ws);

    // 32 * (2000/16) = 4000 waves, 8 waves/block -> 500 blocks
    k_attn_logits<<<dim3(500), dim3(256), 0, stream>>>(x, conv_w, conv_b, ws);

    k_bn_finalize<<<dim3(1), dim3(1), 0, stream>>>(ws, gamma);

    k_softmax<<<dim3(NB), dim3(256), 0, stream>>>(ws, beta);

    // 32 * (1536/16) = 3072 waves, 8 waves/block -> 384 blocks
    k_pool<<<dim3(384), dim3(256), 0, stream>>>(x, ws, out);
}